// PointnetFPModule_66743791780164
// MI455X (gfx1250) — compile-verified
//
#include <hip/hip_runtime.h>
#include <hip/hip_bf16.h>
#include <math.h>

// ---- fixed problem shapes ----
#define BB   16
#define NN   4096
#define MM   1024
#define C1   256
#define C2   512
#define CIN  768
#define HH   256
#define BN_EPS 1e-5f

typedef float v2f __attribute__((ext_vector_type(2)));
typedef float v8f __attribute__((ext_vector_type(8)));

// ----------------------------------------------------------------------------
// Kernel 1: three_nn + inverse-distance weights
// one thread per (b, query point); known xyz staged in LDS (12KB)
// ----------------------------------------------------------------------------
__global__ void fp_three_nn_kernel(const float* __restrict__ unknown,
                                   const float* __restrict__ known,
                                   int* __restrict__ idx_out,
                                   float* __restrict__ w_out) {
    __shared__ float kx[MM], ky[MM], kz[MM];
    const int b = blockIdx.y;
    const int p = blockIdx.x * blockDim.x + threadIdx.x;
    const float* kb = known + (size_t)b * MM * 3;
    for (int j = threadIdx.x; j < MM; j += blockDim.x) {
        kx[j] = kb[j * 3 + 0];
        ky[j] = kb[j * 3 + 1];
        kz[j] = kb[j * 3 + 2];
    }
    __syncthreads();

    const float* up = unknown + ((size_t)b * NN + p) * 3;
    const float ux = up[0], uy = up[1], uz = up[2];

    float d0 = 1e30f, d1 = 1e30f, d2 = 1e30f;
    int   i0 = 0,     i1 = 0,     i2 = 0;
    for (int j = 0; j < MM; ++j) {
        const float dx = ux - kx[j];
        const float dy = uy - ky[j];
        const float dz = uz - kz[j];
        const float d  = dx * dx + dy * dy + dz * dz;
        if (d < d0)      { d2 = d1; i2 = i1; d1 = d0; i1 = i0; d0 = d; i0 = j; }
        else if (d < d1) { d2 = d1; i2 = i1; d1 = d;  i1 = j; }
        else if (d < d2) { d2 = d;  i2 = j; }
    }
    const float r0 = 1.0f / (sqrtf(fmaxf(d0, 0.0f)) + 1e-8f);
    const float r1 = 1.0f / (sqrtf(fmaxf(d1, 0.0f)) + 1e-8f);
    const float r2 = 1.0f / (sqrtf(fmaxf(d2, 0.0f)) + 1e-8f);
    const float inv = 1.0f / (r0 + r1 + r2);

    const size_t o = ((size_t)b * NN + p) * 3;
    idx_out[o + 0] = i0; idx_out[o + 1] = i1; idx_out[o + 2] = i2;
    w_out[o + 0] = r0 * inv; w_out[o + 1] = r1 * inv; w_out[o + 2] = r2 * inv;
}

// ----------------------------------------------------------------------------
// WMMA f32 16x16x4 strip core: one wave computes a 32(rows) x 64(cols) strip
// (two 16-row A tiles reuse every B fragment -> 8 WMMAs per 10 VMEM ops).
// A = W (row-major, ld=ldw), B = feats (row-major [K][P], ld=ldx).
// Fragment layouts per CDNA5 ISA 7.12.2:
//   A 16x4 : lane m = lane%16 ; VGPR v holds K = v + (lane/16)*2
//   B 4x16 : lane n = lane%16 ; VGPR v holds K-row v + (lane/16)*2
//   C/D    : VGPR v, lanes0-15 -> M=v, lanes16-31 -> M=v+8
// BNRELU folds per-channel scale/shift + ReLU into the B-operand load.
// ----------------------------------------------------------------------------
template <bool BNRELU>
__device__ __forceinline__ void wmma_strip32(const float* __restrict__ Wp, int ldw,
                                             const float* __restrict__ Xp, int ldx,
                                             int K, int o0, int p0,
                                             int col, int half,
                                             const float* __restrict__ scl,
                                             const float* __restrict__ sft,
                                             v8f (&acc)[2][4]) {
    for (int k0 = 0; k0 < K; k0 += 4) {
        const int ka = k0 + half * 2;
        const v2f a0 = *(const v2f*)(Wp + (size_t)(o0 + col)      * ldw + ka);
        const v2f a1 = *(const v2f*)(Wp + (size_t)(o0 + 16 + col) * ldw + ka);
        const float* x0 = Xp + (size_t)ka * ldx + p0 + col;
        const float* x1 = x0 + ldx;
        // pull next k-chunk's B lines toward the WGP (global_prefetch_b8;
        // speculative: silently dropped past the end of the buffer)
        __builtin_prefetch(x0 + 4 * (size_t)ldx, 0, 3);
        float s0 = 0.f, t0 = 0.f, s1 = 0.f, t1 = 0.f;
        if (BNRELU) { s0 = scl[ka]; t0 = sft[ka]; s1 = scl[ka + 1]; t1 = sft[ka + 1]; }
#pragma unroll
        for (int t = 0; t < 4; ++t) {
            float bx = x0[t * 16];
            float by = x1[t * 16];
            if (BNRELU) {
                bx = fmaxf(0.0f, s0 * bx + t0);
                by = fmaxf(0.0f, s1 * by + t1);
            }
            v2f bv; bv.x = bx; bv.y = by;
            acc[0][t] = __builtin_amdgcn_wmma_f32_16x16x4_f32(
                false, a0, false, bv, (short)0, acc[0][t], false, false);
            acc[1][t] = __builtin_amdgcn_wmma_f32_16x16x4_f32(
                false, a1, false, bv, (short)0, acc[1][t], false, false);
        }
    }
}

// ----------------------------------------------------------------------------
// Kernel 2: Z = W1[:, :C2] @ known_feats  per batch   (HH x MM per batch)
// ----------------------------------------------------------------------------
__global__ void fp_gemm_z_kernel(const float* __restrict__ W1,
                                 const float* __restrict__ known_feats,
                                 float* __restrict__ Z) {
    const int lane = threadIdx.x & 31, wave = threadIdx.x >> 5;
    const int col = lane & 15, half = lane >> 4;
    const int b  = blockIdx.z;
    const int o0 = blockIdx.y * 32;
    const int p0 = blockIdx.x * 512 + wave * 64;
    const float* Xp = known_feats + (size_t)b * C2 * MM;
    v8f acc[2][4] = {};
    wmma_strip32<false>(W1, CIN, Xp, MM, C2, o0, p0, col, half, nullptr, nullptr, acc);
    float* Zb = Z + (size_t)b * HH * MM;
#pragma unroll
    for (int g = 0; g < 2; ++g) {
#pragma unroll
        for (int t = 0; t < 4; ++t) {
            const int p = p0 + t * 16 + col;
#pragma unroll
            for (int v = 0; v < 8; ++v) {
                const int o = o0 + g * 16 + v + half * 8;
                Zb[(size_t)o * MM + p] = acc[g][t][v];
            }
        }
    }
}

// ----------------------------------------------------------------------------
// Kernel 3: h1_pre = W1[:, C2:] @ unknow_feats + b1 + sum_k w_k * Z[:, idx_k]
// (interpolation commuted past the linear map -> register-level gather)
// ----------------------------------------------------------------------------
__global__ void fp_gemm_h1_kernel(const float* __restrict__ W1,
                                  const float* __restrict__ unknow_feats,
                                  const float* __restrict__ Z,
                                  const int* __restrict__ idx3,
                                  const float* __restrict__ w3,
                                  const float* __restrict__ bias,
                                  float* __restrict__ h1) {
    const int lane = threadIdx.x & 31, wave = threadIdx.x >> 5;
    const int col = lane & 15, half = lane >> 4;
    const int b  = blockIdx.z;
    const int o0 = blockIdx.y * 32;
    const int p0 = blockIdx.x * 512 + wave * 64;
    const float* Xp = unknow_feats + (size_t)b * C1 * NN;
    v8f acc[2][4] = {};
    wmma_strip32<false>(W1 + C2, CIN, Xp, NN, C1, o0, p0, col, half, nullptr, nullptr, acc);

    const float* Zb = Z + (size_t)b * HH * MM;
    float* Hb = h1 + (size_t)b * HH * NN;
#pragma unroll
    for (int t = 0; t < 4; ++t) {
        const int p = p0 + t * 16 + col;
        const size_t q3 = ((size_t)b * NN + p) * 3;
        const int   j0 = idx3[q3 + 0], j1 = idx3[q3 + 1], j2 = idx3[q3 + 2];
        const float u0 = w3[q3 + 0],  u1 = w3[q3 + 1],  u2 = w3[q3 + 2];
#pragma unroll
        for (int g = 0; g < 2; ++g) {
#pragma unroll
            for (int v = 0; v < 8; ++v) {
                const int o = o0 + g * 16 + v + half * 8;
                const float* Zr = Zb + (size_t)o * MM;
                const float val = acc[g][t][v] + bias[o]
                                + u0 * Zr[j0] + u1 * Zr[j1] + u2 * Zr[j2];
                Hb[(size_t)o * NN + p] = val;
            }
        }
    }
}

// ----------------------------------------------------------------------------
// Kernel 4: h2_pre = W2 @ relu(s1*h1_pre + t1) + b2   -> stored into d_out
// BN1+ReLU fused into the B-operand load (s1/t1 cached in LDS)
// ----------------------------------------------------------------------------
__global__ void fp_gemm_h2_kernel(const float* __restrict__ W2,
                                  const float* __restrict__ h1,
                                  const float* __restrict__ s1,
                                  const float* __restrict__ t1,
                                  const float* __restrict__ bias,
                                  float* __restrict__ out) {
    __shared__ float ls[HH], lt[HH];
    for (int i = threadIdx.x; i < HH; i += blockDim.x) { ls[i] = s1[i]; lt[i] = t1[i]; }
    __syncthreads();

    const int lane = threadIdx.x & 31, wave = threadIdx.x >> 5;
    const int col = lane & 15, half = lane >> 4;
    const int b  = blockIdx.z;
    const int o0 = blockIdx.y * 32;
    const int p0 = blockIdx.x * 512 + wave * 64;
    const float* Xp = h1 + (size_t)b * HH * NN;
    v8f acc[2][4] = {};
    wmma_strip32<true>(W2, HH, Xp, NN, HH, o0, p0, col, half, ls, lt, acc);

    float* Ob = out + (size_t)b * HH * NN;
#pragma unroll
    for (int g = 0; g < 2; ++g) {
#pragma unroll
        for (int t = 0; t < 4; ++t) {
            const int p = p0 + t * 16 + col;
#pragma unroll
            for (int v = 0; v < 8; ++v) {
                const int o = o0 + g * 16 + v + half * 8;
                Ob[(size_t)o * NN + p] = acc[g][t][v] + bias[o];
            }
        }
    }
}

// ----------------------------------------------------------------------------
// Deterministic per-channel sum / sum-of-squares over (B, N)
// one block per channel; coalesced strided reads
// ----------------------------------------------------------------------------
__global__ void fp_channel_stats_kernel(const float* __restrict__ x,
                                        float* __restrict__ gsum,
                                        float* __restrict__ gsq) {
    const int c = blockIdx.x;
    float s = 0.f, q = 0.f;
    const int total = BB * NN;  // 65536
    for (int i = threadIdx.x; i < total; i += blockDim.x) {
        const int b = i >> 12;           // / NN
        const int p = i & (NN - 1);
        const float v = x[((size_t)(b * HH + c)) * NN + p];
        s += v; q += v * v;
    }
    __shared__ float ss[256], sq[256];
    ss[threadIdx.x] = s; sq[threadIdx.x] = q;
    __syncthreads();
    for (int st = 128; st > 0; st >>= 1) {
        if ((int)threadIdx.x < st) {
            ss[threadIdx.x] += ss[threadIdx.x + st];
            sq[threadIdx.x] += sq[threadIdx.x + st];
        }
        __syncthreads();
    }
    if (threadIdx.x == 0) { gsum[c] = ss[0]; gsq[c] = sq[0]; }
}

// BN stats -> per-channel scale/shift:  y = s*x + t
__global__ void fp_finalize_stats_kernel(const float* __restrict__ gsum,
                                         const float* __restrict__ gsq,
                                         const float* __restrict__ gamma,
                                         const float* __restrict__ beta,
                                         float* __restrict__ s,
                                         float* __restrict__ t) {
    const int c = threadIdx.x;
    const float inv = 1.0f / (float)(BB * NN);
    const float mean = gsum[c] * inv;
    const float var  = gsq[c] * inv - mean * mean;
    const float sc   = gamma[c] * rsqrtf(var + BN_EPS);
    s[c] = sc;
    t[c] = beta[c] - mean * sc;
}

// Kernel 7: final BN2 + ReLU in place on d_out
__global__ void fp_bn_relu_inplace_kernel(float* __restrict__ y,
                                          const float* __restrict__ s,
                                          const float* __restrict__ t) {
    const size_t total = (size_t)BB * HH * NN;
    for (size_t i = (size_t)blockIdx.x * blockDim.x + threadIdx.x; i < total;
         i += (size_t)gridDim.x * blockDim.x) {
        const int c = (int)((i >> 12) & (HH - 1));  // i / NN % HH
        y[i] = fmaxf(0.0f, s[c] * y[i] + t[c]);
    }
}

// ----------------------------------------------------------------------------
extern "C" void kernel_launch(void* const* d_in, const int* in_sizes, int n_in,
                              void* d_out, int out_size, void* d_ws, size_t ws_size,
                              hipStream_t stream) {
    (void)in_sizes; (void)n_in; (void)out_size; (void)ws_size;
    const float* unknown      = (const float*)d_in[0];
    const float* known        = (const float*)d_in[1];
    const float* unknow_feats = (const float*)d_in[2];
    const float* known_feats  = (const float*)d_in[3];
    const float* W1  = (const float*)d_in[4];
    const float* b1  = (const float*)d_in[5];
    const float* g1  = (const float*)d_in[6];
    const float* be1 = (const float*)d_in[7];
    const float* W2  = (const float*)d_in[8];
    const float* b2  = (const float*)d_in[9];
    const float* g2  = (const float*)d_in[10];
    const float* be2 = (const float*)d_in[11];
    float* out = (float*)d_out;

    // workspace layout (bytes)
    char* ws = (char*)d_ws;
    int*   idxb  = (int*)  (ws);                         // 16*4096*3*4  = 786432
    float* wb    = (float*)(ws + 786432);                // 786432
    float* stats = (float*)(ws + 1572864);               // 8 KB
    float* Z     = (float*)(ws + 1581056);               // 16*256*1024*4 = 16.8 MB
    float* h1    = (float*)(ws + 1581056 + 16777216);    // 16*256*4096*4 = 67.1 MB

    float* sum1 = stats;        float* sq1 = stats + 256;
    float* sum2 = stats + 512;  float* sq2 = stats + 768;
    float* s1   = stats + 1024; float* t1  = stats + 1280;
    float* s2   = stats + 1536; float* t2  = stats + 1792;

    // 1) three_nn + weights
    fp_three_nn_kernel<<<dim3(NN / 256, BB), 256, 0, stream>>>(unknown, known, idxb, wb);

    // 2) Z = W1[:, :C2] @ known_feats     (grid: MM/512, HH/32, B)
    fp_gemm_z_kernel<<<dim3(MM / 512, HH / 32, BB), 256, 0, stream>>>(W1, known_feats, Z);

    // 3) h1_pre = W1[:, C2:] @ unknow_feats + gather(Z) + b1
    fp_gemm_h1_kernel<<<dim3(NN / 512, HH / 32, BB), 256, 0, stream>>>(
        W1, unknow_feats, Z, idxb, wb, b1, h1);

    // 4/5) BN1 stats (deterministic reduction) -> scale/shift
    fp_channel_stats_kernel<<<dim3(HH), 256, 0, stream>>>(h1, sum1, sq1);
    fp_finalize_stats_kernel<<<1, 256, 0, stream>>>(sum1, sq1, g1, be1, s1, t1);

    // 6) h2_pre = W2 @ relu(BN1(h1)) + b2   -> d_out
    fp_gemm_h2_kernel<<<dim3(NN / 512, HH / 32, BB), 256, 0, stream>>>(
        W2, h1, s1, t1, b2, out);

    // 7/8) BN2 stats -> scale/shift
    fp_channel_stats_kernel<<<dim3(HH), 256, 0, stream>>>(out, sum2, sq2);
    fp_finalize_stats_kernel<<<1, 256, 0, stream>>>(sum2, sq2, g2, be2, s2, t2);

    // 9) final BN2 + ReLU in place
    fp_bn_relu_inplace_kernel<<<2048, 256, 0, stream>>>(out, s2, t2);
}